// protein_gnn_28398323761491
// MI455X (gfx1250) — compile-verified
//
#include <hip/hip_runtime.h>

#define N_NODES 50000
#define N_EDGES 600000
#define C 128
#define FE 16
#define NC (N_NODES * C)

typedef float v2f __attribute__((ext_vector_type(2)));
typedef float v8f __attribute__((ext_vector_type(8)));

// ---------------------------------------------------------------------------
// Stage weight matrix W [128 x K] (row-major, row = out-channel) into LDS in
// the exact per-lane B-operand layout for V_WMMA_F32_16X16X4_F32:
//   entry index i = ((kb*8 + ntile)*32 + lane)
//   value        = float2( W[(ntile*16 + lane%16)*K + kb*4 + (lane/16)*2],
//                          W[ ...same row...              + ...      + 1] )
// B-matrix layout (4x16, K x N): VGPR0 lanes0-15 = B[K=0][N=lane], lanes16-31
// = B[K=2][N=lane-16]; VGPR1 = K=1 / K=3.  B[k][n] = W[n][k] (lin = h @ W^T).
// Lane-contiguous float2 entries -> conflict-free ds_load_b64.
// ---------------------------------------------------------------------------
__device__ inline void stage_weights(v2f* sW, const float* __restrict__ W, int K) {
    const int nEntries = (K / 4) * 8 * 32;
    for (int i = threadIdx.x; i < nEntries; i += blockDim.x) {
        int lane = i & 31;
        int nt   = (i >> 5) & 7;
        int kb   = i >> 8;
        int row  = nt * 16 + (lane & 15);
        int col  = kb * 4 + ((lane >> 4) << 1);
        v2f w;
        w.x = W[row * K + col];
        w.y = W[row * K + col + 1];
        sW[i] = w;
    }
}

// ---------------------------------------------------------------------------
// Dual-row-tile accumulate, FULLY UNROLLED over k (K is a compile-time
// constant at every call site after inlining).  No loop phis -> accumulators
// stay pinned in one register set (no v_mov shuffles); the scheduler hoists
// each k-step's 2 global A loads + 4 ds_load_2addr_b64 as early as register
// pressure allows.  Each B fragment feeds two wmma ops.
// A layout (16x4, M x K): lanes0-15 M=lane, K=kb*4+{0,1};
//                         lanes16-31 M=lane-16, K=kb*4+{2,3}.
// ---------------------------------------------------------------------------
__device__ inline void gemm_acc2(v8f acc0[8], v8f acc1[8],
                                 const float* __restrict__ A,
                                 int rowA, int rowB, int K,
                                 const v2f* sW, int lane) {
    const int m  = lane & 15;
    const int kk = (lane >> 4) << 1;
    const float* ar0 = A + (size_t)(rowA + m) * K + kk;
    const float* ar1 = A + (size_t)(rowB + m) * K + kk;
    const int kSteps = K / 4;
#pragma unroll
    for (int kb = 0; kb < kSteps; ++kb) {
        v2f a0 = *(const v2f*)(ar0 + kb * 4);
        v2f a1 = *(const v2f*)(ar1 + kb * 4);
        const v2f* bw = sW + kb * 256 + lane;  // kb*8*32 + lane
        v2f b[8];
#pragma unroll
        for (int t = 0; t < 8; ++t) b[t] = bw[t * 32];
#pragma unroll
        for (int t = 0; t < 8; ++t) {
            acc0[t] = __builtin_amdgcn_wmma_f32_16x16x4_f32(
                false, a0, false, b[t], (short)0, acc0[t], false, false);
            acc1[t] = __builtin_amdgcn_wmma_f32_16x16x4_f32(
                false, a1, false, b[t], (short)0, acc1[t], false, false);
        }
    }
}

// Single-row-tile version (edge kernel, K=16 -> 4 k-steps), fully unrolled.
__device__ inline void gemm_acc1(v8f acc[8], const float* __restrict__ A,
                                 int row0, int K, const v2f* sW, int lane) {
    const int m  = lane & 15;
    const int kk = (lane >> 4) << 1;
    const float* arow = A + (size_t)(row0 + m) * K + kk;
    const int kSteps = K / 4;
#pragma unroll
    for (int kb = 0; kb < kSteps; ++kb) {
        v2f a = *(const v2f*)(arow + kb * 4);
        const v2f* bw = sW + kb * 256 + lane;
        v2f b[8];
#pragma unroll
        for (int t = 0; t < 8; ++t) b[t] = bw[t * 32];
#pragma unroll
        for (int t = 0; t < 8; ++t) {
            acc[t] = __builtin_amdgcn_wmma_f32_16x16x4_f32(
                false, a, false, b[t], (short)0, acc[t], false, false);
        }
    }
}

// Epilogue for one 16x128 accumulator tile.
// C/D layout: VGPR r, lanes0-15 -> (M=r, N=lane); lanes16-31 -> (M=r+8).
__device__ inline void store_tile(const v8f acc[8], int row0,
                                  const float* __restrict__ b1,
                                  const float* __restrict__ b2,
                                  const float* __restrict__ addsrc, int do_relu,
                                  float* __restrict__ out, int lane) {
    const int mofs  = (lane & 16) ? 8 : 0;
    const int cbase = lane & 15;
#pragma unroll
    for (int t = 0; t < 8; ++t) {
        const int col = t * 16 + cbase;
        float bias = b1[col] + (b2 ? b2[col] : 0.0f);
#pragma unroll
        for (int r = 0; r < 8; ++r) {
            const int row = row0 + r + mofs;
            float v = acc[t][r] + bias;
            if (do_relu) v = fmaxf(v, 0.0f);
            if (addsrc) v += addsrc[(size_t)row * C + col];
            out[(size_t)row * C + col] = v;
        }
    }
}

// ---------------------------------------------------------------------------
// Generic fused linear tile kernel (32 rows per wave, 4 waves per block):
//   out[row, :] = act( A1@W1^T + b1 [+ A2@W2^T + b2] ) [+ addsrc[row, :]]
// relu applied BEFORE addsrc (matches h = relu(lin(t1)) + x2).  K fixed = 128.
// ---------------------------------------------------------------------------
__global__ __launch_bounds__(128)
void lin_tile_kernel(const float* __restrict__ A1, const float* __restrict__ W1,
                     const float* __restrict__ b1,
                     const float* __restrict__ A2, const float* __restrict__ W2,
                     const float* __restrict__ b2,
                     const float* __restrict__ addsrc, int do_relu,
                     float* __restrict__ out, int nrows) {
    __shared__ v2f sW[(C / 4) * 8 * 32];  // 8192 entries = 64 KB
    const int lane = threadIdx.x & 31;
    const int wave = threadIdx.x >> 5;
    const int tile = blockIdx.x * 4 + wave;
    const int row0 = tile * 32;
    const bool active0 = row0 < nrows;           // wave-uniform -> EXEC all-1s
    const bool active1 = (row0 + 16) < nrows;
    // Partial tile: clamp second sub-tile's rows onto the first (valid memory,
    // results discarded at store).
    const int rowB = active1 ? (row0 + 16) : row0;

    v8f acc0[8] = {};
    v8f acc1[8] = {};

    stage_weights(sW, W1, C);
    __syncthreads();
    if (active0) gemm_acc2(acc0, acc1, A1, row0, rowB, C, sW, lane);

    if (A2 != nullptr) {
        __syncthreads();
        stage_weights(sW, W2, C);
        __syncthreads();
        if (active0) gemm_acc2(acc0, acc1, A2, row0, rowB, C, sW, lane);
    }

    if (active0) store_tile(acc0, row0, b1, b2, addsrc, do_relu, out, lane);
    if (active1) store_tile(acc1, row0 + 16, b1, b2, addsrc, do_relu, out, lane);
}

// ---------------------------------------------------------------------------
// Edge kernel: ew = fe @ We1^T + be1  (K=16 WMMA), then
//   msg = ew * x1[src], atomic scatter-add into agg[dst]; degree counts folded
//   into the same epilogue (lanes with cbase==0 cover each edge exactly once).
// ---------------------------------------------------------------------------
__global__ __launch_bounds__(128)
void edge_kernel(const float* __restrict__ fe, const float* __restrict__ We1,
                 const float* __restrict__ be1,
                 const int* __restrict__ src, const int* __restrict__ dst,
                 const float* __restrict__ x1, float* __restrict__ agg,
                 float* __restrict__ cnt) {
    __shared__ v2f sW[(FE / 4) * 8 * 32];  // 1024 entries = 8 KB
    stage_weights(sW, We1, FE);
    __syncthreads();

    const int lane = threadIdx.x & 31;
    const int wave = threadIdx.x >> 5;
    const int tile = blockIdx.x * 4 + wave;
    const int e0 = tile * 16;
    if (e0 >= N_EDGES) return;

    v8f acc[8] = {};
    gemm_acc1(acc, fe, e0, FE, sW, lane);

    const int mofs  = (lane & 16) ? 8 : 0;
    const int cbase = lane & 15;
#pragma unroll
    for (int r = 0; r < 8; ++r) {
        const int e = e0 + r + mofs;
        const int s = src[e];
        const int d = dst[e];
        const float* xs = x1 + (size_t)s * C;
        float* ad = agg + (size_t)d * C;
        if (cbase == 0) unsafeAtomicAdd(&cnt[d], 1.0f);  // one per edge
#pragma unroll
        for (int t = 0; t < 8; ++t) {
            const int col = t * 16 + cbase;
            const float ew = acc[t][r] + be1[col];
            unsafeAtomicAdd(&ad[col], ew * xs[col]);  // global_atomic_add_f32
        }
    }
}

__global__ void zero_kernel(float* __restrict__ p, int n) {
    const int i = blockIdx.x * blockDim.x + threadIdx.x;
    if (i < n) p[i] = 0.0f;
}

__global__ void finalize_kernel(float* __restrict__ agg, const float* __restrict__ cnt) {
    const int i = blockIdx.x * blockDim.x + threadIdx.x;
    if (i < NC) {
        const float c = cnt[i >> 7];  // i / C
        agg[i] = agg[i] / fmaxf(c, 1.0f);
    }
}

// ---------------------------------------------------------------------------
extern "C" void kernel_launch(void* const* d_in, const int* in_sizes, int n_in,
                              void* d_out, int out_size, void* d_ws, size_t ws_size,
                              hipStream_t stream) {
    const float* x   = (const float*)d_in[0];
    const float* fe  = (const float*)d_in[1];
    const int*   ei  = (const int*)d_in[2];
    const float* Wx1 = (const float*)d_in[3];  const float* bx1 = (const float*)d_in[4];
    const float* Wx2 = (const float*)d_in[5];  const float* bx2 = (const float*)d_in[6];
    const float* We1 = (const float*)d_in[7];  const float* be1 = (const float*)d_in[8];
    const float* We2 = (const float*)d_in[9];  const float* be2 = (const float*)d_in[10];
    const float* Wl  = (const float*)d_in[11]; const float* bl  = (const float*)d_in[12];
    const float* Wr  = (const float*)d_in[13]; const float* br  = (const float*)d_in[14];
    const float* Wf  = (const float*)d_in[15]; const float* bf  = (const float*)d_in[16];

    const int* src = ei;            // t_edge_index[0]
    const int* dst = ei + N_EDGES;  // t_edge_index[1]

    float* ws  = (float*)d_ws;
    float* x1  = ws;            // [N, C]
    float* x2  = ws + NC;       // [N, C]
    float* buf = ws + 2 * NC;   // [N, C]  agg -> t1 -> h (in-place, per-row safe)
    float* cnt = ws + 3 * NC;   // [N]

    const int nodeTiles  = (N_NODES + 31) / 32;     // 1563 (32 rows per wave)
    const int nodeBlocks = (nodeTiles + 3) / 4;     // 391
    const int edgeTiles  = N_EDGES / 16;            // 37500
    const int edgeBlocks = (edgeTiles + 3) / 4;     // 9375

    // zero agg + cnt (contiguous)
    {
        const int n = NC + N_NODES;
        zero_kernel<<<(n + 255) / 256, 256, 0, stream>>>(buf, n);
    }

    // x1 = relu(x @ Wx1^T + bx1); x2 = relu(x @ Wx2^T + bx2)
    lin_tile_kernel<<<nodeBlocks, 128, 0, stream>>>(
        x, Wx1, bx1, nullptr, nullptr, nullptr, nullptr, 1, x1, N_NODES);
    lin_tile_kernel<<<nodeBlocks, 128, 0, stream>>>(
        x, Wx2, bx2, nullptr, nullptr, nullptr, nullptr, 1, x2, N_NODES);

    // edge message + scatter-add + degree counts, then scatter-mean
    edge_kernel<<<edgeBlocks, 128, 0, stream>>>(fe, We1, be1, src, dst, x1, buf, cnt);
    finalize_kernel<<<(NC + 255) / 256, 256, 0, stream>>>(buf, cnt);

    // t1 = agg @ Wl^T + bl + x1 @ Wr^T + br      (in-place into buf)
    lin_tile_kernel<<<nodeBlocks, 128, 0, stream>>>(
        buf, Wl, bl, x1, Wr, br, nullptr, 0, buf, N_NODES);

    // h = relu(t1 @ We2^T + be2) + x2            (in-place into buf)
    lin_tile_kernel<<<nodeBlocks, 128, 0, stream>>>(
        buf, We2, be2, nullptr, nullptr, nullptr, x2, 1, buf, N_NODES);

    // out = h @ Wf^T + bf
    lin_tile_kernel<<<nodeBlocks, 128, 0, stream>>>(
        buf, Wf, bf, nullptr, nullptr, nullptr, nullptr, 0, (float*)d_out, N_NODES);
}